// StateNetwork_70282844832098
// MI455X (gfx1250) — compile-verified
//
#include <hip/hip_runtime.h>
#include <cstdint>
#include <cstddef>

#define NN    2048
#define EMB   128
#define NHID  3
#define BB    32
#define ALPHA 0.2f
#define FILLV 9e-15f

typedef float v2f __attribute__((ext_vector_type(2)));
typedef float v8f __attribute__((ext_vector_type(8)));

// workspace layout (in floats)
#define WS_H    0              // 2048*3
#define WS_SSRC 6144           // 2048
#define WS_SDST 8192           // 2048
#define WS_HTOT 10240          // 3 (+pad)
#define WS_HP   10248          // 32*6144 (A-matrix for final FC), 8B-aligned

__device__ __forceinline__ float lrelu(float x) { return x > 0.0f ? x : ALPHA * x; }

// ---------------------------------------------------------------------------
// Kernel 1: h = ent_emb @ W ; s_src = h@a[:3] ; s_dst = h@a[3:]
// one wave per node row, shuffle reduce
// ---------------------------------------------------------------------------
__global__ void __launch_bounds__(256) k_prep(const float* __restrict__ ent,
                                              const float* __restrict__ W,
                                              const float* __restrict__ a,
                                              float* __restrict__ ws) {
  int lane = threadIdx.x & 31;
  int wave = threadIdx.x >> 5;
  int i = blockIdx.x * 8 + wave;
  const float* row = ent + (size_t)i * EMB;
  float h0 = 0.f, h1 = 0.f, h2 = 0.f;
#pragma unroll
  for (int t = 0; t < 4; ++t) {
    int e = t * 32 + lane;
    float x = row[e];
    h0 = fmaf(x, W[e * 3 + 0], h0);
    h1 = fmaf(x, W[e * 3 + 1], h1);
    h2 = fmaf(x, W[e * 3 + 2], h2);
  }
#pragma unroll
  for (int s = 16; s; s >>= 1) {
    h0 += __shfl_xor(h0, s, 32);
    h1 += __shfl_xor(h1, s, 32);
    h2 += __shfl_xor(h2, s, 32);
  }
  if (lane == 0) {
    ws[WS_H + i * 3 + 0] = h0;
    ws[WS_H + i * 3 + 1] = h1;
    ws[WS_H + i * 3 + 2] = h2;
    ws[WS_SSRC + i] = h0 * a[0] + h1 * a[1] + h2 * a[2];
    ws[WS_SDST + i] = h0 * a[3] + h1 * a[4] + h2 * a[5];
  }
}

// ---------------------------------------------------------------------------
// Kernel 1b: Htot[k] = sum_j h[j][k]   (single block, deterministic)
// ---------------------------------------------------------------------------
__global__ void __launch_bounds__(256) k_htot(float* __restrict__ ws) {
  __shared__ float red[8][3];
  float s0 = 0.f, s1 = 0.f, s2 = 0.f;
  for (int i = threadIdx.x; i < NN; i += 256) {
    s0 += ws[WS_H + i * 3 + 0];
    s1 += ws[WS_H + i * 3 + 1];
    s2 += ws[WS_H + i * 3 + 2];
  }
#pragma unroll
  for (int s = 16; s; s >>= 1) {
    s0 += __shfl_xor(s0, s, 32);
    s1 += __shfl_xor(s1, s, 32);
    s2 += __shfl_xor(s2, s, 32);
  }
  int lane = threadIdx.x & 31, wave = threadIdx.x >> 5;
  if (lane == 0) { red[wave][0] = s0; red[wave][1] = s1; red[wave][2] = s2; }
  __syncthreads();
  if (threadIdx.x == 0) {
    float t0 = 0.f, t1 = 0.f, t2 = 0.f;
    for (int w = 0; w < 8; ++w) { t0 += red[w][0]; t1 += red[w][1]; t2 += red[w][2]; }
    ws[WS_HTOT + 0] = t0; ws[WS_HTOT + 1] = t1; ws[WS_HTOT + 2] = t2;
  }
}

// ---------------------------------------------------------------------------
// Kernel 2: streaming masked softmax + att@h + ELU  (memory-bound, 512 MB)
// one wave per adjacency row; single global pass, mask replay from registers
// ---------------------------------------------------------------------------
__global__ void __launch_bounds__(256) k_gat(const int* __restrict__ adj,
                                             float* __restrict__ ws) {
  __shared__ float sh_h[NN * 3];   // 24 KB
  __shared__ float sh_sd[NN];      //  8 KB

  {
    const float4* gh = (const float4*)(ws + WS_H);
    float4* lh = (float4*)sh_h;
    for (int t = threadIdx.x; t < (NN * 3) / 4; t += 256) lh[t] = gh[t];
    const float4* gs = (const float4*)(ws + WS_SDST);
    float4* ls = (float4*)sh_sd;
    for (int t = threadIdx.x; t < NN / 4; t += 256) ls[t] = gs[t];
  }
  __syncthreads();

  float ht0 = ws[WS_HTOT + 0], ht1 = ws[WS_HTOT + 1], ht2 = ws[WS_HTOT + 2];
  int lane = threadIdx.x & 31;
  int wave = threadIdx.x >> 5;

  for (int q = 0; q < 4; ++q) {
    int r = blockIdx.x * 32 + wave * 4 + q;   // global row id, 0..B*N-1
    int b = r >> 11;
    int i = r & (NN - 1);
    float s_i = ws[WS_SSRC + i];
    const int4* arow = (const int4*)(adj + ((size_t)b * NN + i) * NN);

    // ---- pass 1: stream adjacency once, record mask bits in registers ----
    unsigned long long mask = 0ull;
    float m = -__builtin_inff();
    int cnt = 0;
#pragma unroll
    for (int it = 0; it < 16; ++it) {
      int4 av = arow[it * 32 + lane];
      int jb = it * 128 + lane * 4;
      float e0 = lrelu(s_i + sh_sd[jb + 0]);
      float e1 = lrelu(s_i + sh_sd[jb + 1]);
      float e2 = lrelu(s_i + sh_sd[jb + 2]);
      float e3 = lrelu(s_i + sh_sd[jb + 3]);
      unsigned b0 = av.x > 0, b1 = av.y > 0, b2 = av.z > 0, b3 = av.w > 0;
      m = fmaxf(m, b0 ? e0 : -__builtin_inff());
      m = fmaxf(m, b1 ? e1 : -__builtin_inff());
      m = fmaxf(m, b2 ? e2 : -__builtin_inff());
      m = fmaxf(m, b3 ? e3 : -__builtin_inff());
      cnt += (int)(b0 + b1 + b2 + b3);
      mask |= (unsigned long long)(b0 | (b1 << 1) | (b2 << 2) | (b3 << 3)) << (it * 4);
    }
#pragma unroll
    for (int s = 16; s; s >>= 1) {
      m = fmaxf(m, __shfl_xor(m, s, 32));
      cnt += __shfl_xor(cnt, s, 32);
    }
    float mf = (cnt < NN) ? fmaxf(m, FILLV) : m;   // masked entries join the max

    // ---- pass 2: replay from register mask (no global re-read) ----
    float d = 0.f, v0 = 0.f, v1 = 0.f, v2 = 0.f, u0 = 0.f, u1 = 0.f, u2 = 0.f;
#pragma unroll
    for (int it = 0; it < 16; ++it) {
      unsigned bits = (unsigned)(mask >> (it * 4)) & 0xFu;
      int jb = it * 128 + lane * 4;
#pragma unroll
      for (int c = 0; c < 4; ++c) {
        float t = (float)((bits >> c) & 1u);       // branch-free: keep wave dense
        int j = jb + c;
        float e = lrelu(s_i + sh_sd[j]);
        float w = __expf(e - mf) * t;
        float g0 = sh_h[j * 3 + 0], g1 = sh_h[j * 3 + 1], g2 = sh_h[j * 3 + 2];
        d += w;
        u0 = fmaf(t, g0, u0); u1 = fmaf(t, g1, u1); u2 = fmaf(t, g2, u2);
        v0 = fmaf(w, g0, v0); v1 = fmaf(w, g1, v1); v2 = fmaf(w, g2, v2);
      }
    }
#pragma unroll
    for (int s = 16; s; s >>= 1) {
      d  += __shfl_xor(d,  s, 32);
      v0 += __shfl_xor(v0, s, 32);
      v1 += __shfl_xor(v1, s, 32);
      v2 += __shfl_xor(v2, s, 32);
      u0 += __shfl_xor(u0, s, 32);
      u1 += __shfl_xor(u1, s, 32);
      u2 += __shfl_xor(u2, s, 32);
    }
    if (lane == 0) {
      // closed-form bulk contribution of all masked (FILL-valued) entries
      float em = __expf(FILLV - mf);
      float denom = d + (float)(NN - cnt) * em;
      float inv = 1.0f / denom;
      float p0 = (v0 + em * (ht0 - u0)) * inv;
      float p1 = (v1 + em * (ht1 - u1)) * inv;
      float p2 = (v2 + em * (ht2 - u2)) * inv;
      p0 = p0 > 0.f ? p0 : (__expf(p0) - 1.0f);   // ELU
      p1 = p1 > 0.f ? p1 : (__expf(p1) - 1.0f);
      p2 = p2 > 0.f ? p2 : (__expf(p2) - 1.0f);
      size_t o = (size_t)WS_HP + (size_t)b * (NN * NHID) + (size_t)i * 3;
      ws[o + 0] = p0; ws[o + 1] = p1; ws[o + 2] = p2;
    }
  }
}

// ---------------------------------------------------------------------------
// Kernel 3: out[32,128] = hp[32,6144] @ fc1_w[128,6144]^T + b  via f32 WMMA
// 16 output tiles of 16x16, one wave each; 4 independent accumulators
// ---------------------------------------------------------------------------
__global__ void __launch_bounds__(256) k_fc(const float* __restrict__ ws,
                                            const float* __restrict__ fw,
                                            const float* __restrict__ fb,
                                            float* __restrict__ out) {
  int lane = threadIdx.x & 31;
  int wave = threadIdx.x >> 5;
  int tile = blockIdx.x * 8 + wave;      // 0..15
  int tm = tile >> 3;                    // 0..1 (rows of 16 in B=32)
  int tn = tile & 7;                     // 0..7 (cols of 16 in EMB=128)
  int rsel = lane & 15;
  int kb = (lane >> 4) << 1;             // lanes 0-15 -> K0,K1 ; lanes 16-31 -> K2,K3

  const float* arow = ws + WS_HP + (size_t)(tm * 16 + rsel) * 6144 + kb;
  const float* brow = fw + (size_t)(tn * 16 + rsel) * 6144 + kb;

  v8f a0 = {0.f,0.f,0.f,0.f,0.f,0.f,0.f,0.f};
  v8f a1 = a0, a2 = a0, a3 = a0;
  for (int k0 = 0; k0 < 6144; k0 += 16) {
    v2f A0 = *(const v2f*)(arow + k0 + 0);
    v2f B0 = *(const v2f*)(brow + k0 + 0);
    a0 = __builtin_amdgcn_wmma_f32_16x16x4_f32(false, A0, false, B0, (short)0, a0, false, false);
    v2f A1 = *(const v2f*)(arow + k0 + 4);
    v2f B1 = *(const v2f*)(brow + k0 + 4);
    a1 = __builtin_amdgcn_wmma_f32_16x16x4_f32(false, A1, false, B1, (short)0, a1, false, false);
    v2f A2 = *(const v2f*)(arow + k0 + 8);
    v2f B2 = *(const v2f*)(brow + k0 + 8);
    a2 = __builtin_amdgcn_wmma_f32_16x16x4_f32(false, A2, false, B2, (short)0, a2, false, false);
    v2f A3 = *(const v2f*)(arow + k0 + 12);
    v2f B3 = *(const v2f*)(brow + k0 + 12);
    a3 = __builtin_amdgcn_wmma_f32_16x16x4_f32(false, A3, false, B3, (short)0, a3, false, false);
  }
  v8f acc = (a0 + a1) + (a2 + a3);

  int col = tn * 16 + rsel;
  float bias = fb[col];
  int rbase = tm * 16 + ((lane >> 4) << 3);   // D layout: lanes 16-31 hold M=r+8
#pragma unroll
  for (int r2 = 0; r2 < 8; ++r2) {
    out[(size_t)(rbase + r2) * EMB + col] = acc[r2] + bias;
  }
}

// ---------------------------------------------------------------------------
extern "C" void kernel_launch(void* const* d_in, const int* in_sizes, int n_in,
                              void* d_out, int out_size, void* d_ws, size_t ws_size,
                              hipStream_t stream) {
  const int*   graph = (const int*)d_in[0];    // [32,2048,2048] int32
  const float* ent   = (const float*)d_in[1];  // [2048,128]
  const float* W     = (const float*)d_in[2];  // [128,3]
  const float* a     = (const float*)d_in[3];  // [6,1]
  const float* fw    = (const float*)d_in[4];  // [128,6144]
  const float* fb    = (const float*)d_in[5];  // [128]
  float* ws  = (float*)d_ws;
  float* out = (float*)d_out;

  k_prep<<<NN / 8, 256, 0, stream>>>(ent, W, a, ws);
  k_htot<<<1, 256, 0, stream>>>(ws);
  k_gat<<<(BB * NN) / 32, 256, 0, stream>>>(graph, ws);
  k_fc<<<2, 256, 0, stream>>>(ws, fw, fb, out);
}